// FeedForwardQuantum_65481071405813
// MI455X (gfx1250) — compile-verified
//
#include <hip/hip_runtime.h>
#include <math.h>

typedef float v2f __attribute__((ext_vector_type(2)));
typedef float v8f __attribute__((ext_vector_type(8)));

#define EMBED 512
#define FFN   64
#define NQ    16

#define TILES_PER_WAVE  2
#define WAVES_PER_BLOCK 8
#define TILES_PER_BLOCK (TILES_PER_WAVE * WAVES_PER_BLOCK)   // 16 tiles = 256 rows / block
#define LDS_STRIDE 68                                        // padded: bank-conflict-free both ways
#define TILE_LDS   (16 * LDS_STRIDE)

// D = A(16x4 f32) * B(4x16 f32) + C(16x16 f32), exact fp32 matrix op
static __device__ __forceinline__ v8f wmma4(v2f a, v2f b, v8f c) {
    return __builtin_amdgcn_wmma_f32_16x16x4_f32(false, a, false, b, (short)0, c, false, false);
}

__global__ __launch_bounds__(256)
void ffnq_kernel(const float* __restrict__ x,
                 const float* __restrict__ theta,
                 const float* __restrict__ W1,   // [64,16]
                 const float* __restrict__ b1,   // [64]
                 const float* __restrict__ W2,   // [512,64]
                 const float* __restrict__ b2,   // [512]
                 float* __restrict__ out)        // [rows,512]
{
    __shared__ float smem[TILES_PER_BLOCK * TILE_LDS];

    const int tid  = threadIdx.x;
    const int wave = tid >> 5;
    const int lane = tid & 31;
    const int nl   = lane & 15;       // N (or M for A-frags) within 16-tile
    const int hi   = lane >> 4;       // half-wave select
    const int koff = hi * 2;          // A/B frags: upper lanes carry K=2,3 of each chunk
    const int mOff = hi * 8;          // C/D frags: upper lanes carry M=8..15

    // ---------------- Phase 1: Z -> GEMM1 -> ReLU -> H staged in LDS ----------------
    for (int t = 0; t < TILES_PER_WAVE; ++t) {
        const int slot = wave * TILES_PER_WAVE + t;
        const int tile = blockIdx.x * TILES_PER_BLOCK + slot;
        const int row  = tile * 16 + nl;

        // Z in f32 A-matrix layout: 4 K-chunks of 4
        v2f Az[4];
#pragma unroll
        for (int c = 0; c < 4; ++c) {
            const int k0 = 4 * c + koff;
            v2f xin = *(const v2f*)(x + row * EMBED + k0);
            Az[c].x = cosf(xin.x) * cosf(theta[k0]);
            Az[c].y = cosf(xin.y) * cosf(theta[k0 + 1]);
        }

        // GEMM1: H[16,64] = Z[16,16] x W1^T, 4 N-tiles x 4 K-chunks
#pragma unroll
        for (int n = 0; n < 4; ++n) {
            const float bias = b1[16 * n + nl];
            v8f acc;
#pragma unroll
            for (int i = 0; i < 8; ++i) acc[i] = bias;
#pragma unroll
            for (int c = 0; c < 4; ++c) {
                v2f Bw = *(const v2f*)(W1 + (16 * n + nl) * NQ + 4 * c + koff);
                acc = wmma4(Az[c], Bw, acc);
            }
            // ReLU + stash row-major (padded stride) into LDS
#pragma unroll
            for (int i = 0; i < 8; ++i) {
                float h = acc[i] > 0.0f ? acc[i] : 0.0f;
                smem[slot * TILE_LDS + (i + mOff) * LDS_STRIDE + 16 * n + nl] = h;
            }
        }
    }

    // Same-wave LDS RAW: DS ops are in-order per wave; make the wait explicit.
    asm volatile("s_wait_dscnt 0" ::: "memory");

    // ---------------- Phase 2: A2 = H (transposed via LDS), GEMM2 over 32 embed tiles ----
    v2f A2[TILES_PER_WAVE][16];
#pragma unroll
    for (int t = 0; t < TILES_PER_WAVE; ++t) {
        const int slot = wave * TILES_PER_WAVE + t;
#pragma unroll
        for (int c = 0; c < 16; ++c)
            A2[t][c] = *(const v2f*)&smem[slot * TILE_LDS + nl * LDS_STRIDE + 4 * c + koff];
    }

    const int rowBase0 = (blockIdx.x * TILES_PER_BLOCK + wave * TILES_PER_WAVE) * 16;

    for (int n2 = 0; n2 < 32; ++n2) {
        const int e = 16 * n2 + nl;
        const float bias = b2[e];
        v8f acc0, acc1;
#pragma unroll
        for (int i = 0; i < 8; ++i) { acc0[i] = bias; acc1[i] = bias; }

#pragma unroll
        for (int c = 0; c < 16; ++c) {
            v2f Bw = *(const v2f*)(W2 + e * FFN + 4 * c + koff);  // shared by both M-tiles
            acc0 = wmma4(A2[0][c], Bw, acc0);
            acc1 = wmma4(A2[1][c], Bw, acc1);
        }

        if (n2 + 1 < 32)
            __builtin_prefetch(W2 + (16 * (n2 + 1) + nl) * FFN, 0, 0);  // global_prefetch_b8

        // coalesced stores: each half-wave writes a 64B run per row
#pragma unroll
        for (int i = 0; i < 8; ++i) {
            out[(rowBase0 + i + mOff) * EMBED + e]      = acc0[i];
            out[(rowBase0 + 16 + i + mOff) * EMBED + e] = acc1[i];
        }
    }
}

extern "C" void kernel_launch(void* const* d_in, const int* in_sizes, int n_in,
                              void* d_out, int out_size, void* d_ws, size_t ws_size,
                              hipStream_t stream) {
    const float* x     = (const float*)d_in[0];
    const float* theta = (const float*)d_in[1];
    const float* W1    = (const float*)d_in[2];
    const float* b1    = (const float*)d_in[3];
    const float* W2    = (const float*)d_in[4];
    const float* b2    = (const float*)d_in[5];
    float* out = (float*)d_out;

    const int rows   = in_sizes[0] / EMBED;              // 65536
    const int blocks = rows / (16 * TILES_PER_BLOCK);    // 256

    ffnq_kernel<<<blocks, 256, 0, stream>>>(x, theta, W1, b1, W2, b2, out);
}